// Tagger_88923002896448
// MI455X (gfx1250) — compile-verified
//
#include <hip/hip_runtime.h>

// HMM forward (logZ) for B=1024, T=256, N=64 on gfx1250.
// Scaled linear-space forward recurrence; per 16-batch tile per timestep:
//   s[16x64] = U_old[16x64] @ trans[64x64]          (16 chained f32 WMMAs)
//   U_new[row,j] = mask ? s[row,j] * inv_rowmax[row] * emit[t,row,j] : U_old[row,j]
//   C2[row]     += mask ? log2(rowmax[row]) : 0     (per-row log2 scale)
// 64 workgroups x 128 threads (4 waves); wave w owns output states [16w,16w+16).
// Alpha exchanged via double-buffered LDS (row pad 68 -> conflict-free b64 loads),
// one s_barrier per timestep. Row max computed from the A-operand values each
// lane already loads (half-row) + one shfl_xor with the partner half-lane.

#define B_SZ 1024
#define T_SZ 256
#define N_SZ 64
#define NGROUPS (B_SZ / 16)
#define LN2F 0.69314718055994530942f

typedef float v2f __attribute__((ext_vector_type(2)));
typedef float v8f __attribute__((ext_vector_type(8)));

__global__ __launch_bounds__(128) void hmm_fwd_kernel(
    const float* __restrict__ emit,          // [B,T,N]
    const unsigned char* __restrict__ mask,  // [B,T] bool (1 byte)
    const float* __restrict__ trans,         // [N,N]
    const float* __restrict__ strans,        // [N]
    const float* __restrict__ etrans,        // [N]
    float* __restrict__ partial)             // [NGROUPS] per-group logZ sums (ln)
{
  __shared__ float U[2][16][68];  // double-buffered unscaled alpha
  __shared__ float PS[4][16];     // final per-wave partial row dot-products

  const int tid  = threadIdx.x;
  const int wid  = tid >> 5;        // wave 0..3 -> output-state tile
  const int lane = tid & 31;
  const int half = lane >> 4;       // A/B K-offset half, C/D row half
  const int arow = lane & 15;       // batch row this lane feeds into A
  const int col  = wid * 16 + arow; // output state column this lane holds in C/D
  const int bbase = blockIdx.x * 16;

  // B-operand (trans) resident in VGPRs. Chunk c covers K = 4c..4c+3.
  // B 4x16 layout mirrors documented A 16x4: VGPR0 = K0 (lanes 0-15) / K2
  // (lanes 16-31); VGPR1 = K1 / K3.
  float bt0[16], bt1[16];
  #pragma unroll
  for (int c = 0; c < 16; ++c) {
    const int k0 = 4 * c + (half ? 2 : 0);
    bt0[c] = trans[k0 * N_SZ + col];
    bt1[c] = trans[(k0 + 1) * N_SZ + col];
  }
  const float etr = etrans[col];

  // t = 0: alpha0 = strans * emit[:,0,:]  (linear space, C2 = 0)
  float prevU[8];       // this wave's unscaled alpha slice, D-layout
  float Crow2 = 0.0f;   // log2-scale for batch row `arow` (replicated per wave)
  #pragma unroll
  for (int r = 0; r < 8; ++r) {
    const int dr = r + (half ? 8 : 0);
    const float u0 = strans[col] * emit[(size_t)(bbase + dr) * (T_SZ * N_SZ) + col];
    prevU[r] = u0;
    U[0][dr][col] = u0;
  }
  __syncthreads();

  for (int t = 1; t < T_SZ; ++t) {
    const int pr = (t - 1) & 1, pw = t & 1;

    // Load this lane's half of its batch row (A-operand, raw/unscaled).
    v2f a[16];
    #pragma unroll
    for (int c = 0; c < 16; ++c)
      a[c] = *(const v2f*)&U[pr][arow][4 * c + (half ? 2 : 0)];

    // s = U_old @ trans_tile: 16 chained f32 WMMAs (starts as soon as loads land).
    v8f acc = {};
    #pragma unroll
    for (int c = 0; c < 16; ++c) {
      v2f b; b.x = bt0[c]; b.y = bt1[c];
      acc = __builtin_amdgcn_wmma_f32_16x16x4_f32(
          /*neg_a=*/false, a[c], /*neg_b=*/false, b,
          /*c_mod=*/(short)0, acc, /*reuse_a=*/false, /*reuse_b=*/false);
    }

    // Row max: fmax tree over the 32 values already in registers (half row),
    // then combine with the partner half-lane. Overlaps with the WMMA chain.
    float hm = fmaxf(a[0].x, a[0].y);
    #pragma unroll
    for (int c = 1; c < 16; ++c) hm = fmaxf(hm, fmaxf(a[c].x, a[c].y));
    const float m0  = fmaxf(hm, __shfl_xor(hm, 16, 32));
    const float inv = __builtin_amdgcn_rcpf(m0);       // v_rcp_f32 (1 ulp)
    const float l2  = __builtin_amdgcn_logf(m0);       // v_log_f32 = log2
    const unsigned char mka = mask[(bbase + arow) * T_SZ + t];
    Crow2 += mka ? l2 : 0.0f;  // per-row scale only advances when row is active

    // Blend with mask (frozen rows keep U and C unchanged), store new alpha.
    #pragma unroll
    for (int r = 0; r < 8; ++r) {
      const int dr = r + (half ? 8 : 0);
      const float invd = __shfl(inv, dr, 32);  // inv_rowmax for this D-row
      const size_t eoff =
          (size_t)(bbase + dr) * (T_SZ * N_SZ) + (size_t)t * N_SZ + col;
      const float me = emit[eoff];
      const unsigned char mk = mask[(bbase + dr) * T_SZ + t];
      if (t + 1 < T_SZ) __builtin_prefetch(&emit[eoff + N_SZ], 0, 1);
      const float un = mk ? acc[r] * (invd * me) : prevU[r];
      prevU[r] = un;
      U[pw][dr][col] = un;
    }
    __syncthreads();
  }

  // logZ_row = ln(sum_j U[row,j]*etrans[j]) + C2[row]*ln2; sum 16 rows per group.
  #pragma unroll
  for (int r = 0; r < 8; ++r) {
    const int dr = r + (half ? 8 : 0);
    float v = prevU[r] * etr;
    #pragma unroll
    for (int m = 1; m < 16; m <<= 1) v += __shfl_xor(v, m, 16);
    if (arow == r) PS[wid][dr] = v;
  }
  __syncthreads();
  if (wid == 0 && lane < 16) {
    const float tot = PS[0][lane] + PS[1][lane] + PS[2][lane] + PS[3][lane];
    float lz = (__builtin_amdgcn_logf(tot) + Crow2) * LN2F;
    #pragma unroll
    for (int m = 1; m < 16; m <<= 1) lz += __shfl_xor(lz, m, 16);
    if (lane == 0) partial[blockIdx.x] = lz;
  }
}

// Deterministic fixed-order reduction of the 64 group partials; out = sum / B.
__global__ void hmm_finalize_kernel(const float* __restrict__ partial,
                                    float* __restrict__ out) {
  if (threadIdx.x == 0 && blockIdx.x == 0) {
    double s = 0.0;
    for (int i = 0; i < NGROUPS; ++i) s += (double)partial[i];
    out[0] = (float)(s / (double)B_SZ);
  }
}

extern "C" void kernel_launch(void* const* d_in, const int* in_sizes, int n_in,
                              void* d_out, int out_size, void* d_ws, size_t ws_size,
                              hipStream_t stream) {
  const float* emit         = (const float*)d_in[0];
  const unsigned char* mask = (const unsigned char*)d_in[1];  // jnp bool = 1 byte
  const float* trans        = (const float*)d_in[2];
  const float* strans       = (const float*)d_in[3];
  const float* etrans       = (const float*)d_in[4];
  float* out                = (float*)d_out;
  float* partial            = (float*)d_ws;  // NGROUPS floats

  hmm_fwd_kernel<<<NGROUPS, 128, 0, stream>>>(emit, mask, trans, strans, etrans,
                                              partial);
  hmm_finalize_kernel<<<1, 32, 0, stream>>>(partial, out);
}